// BioBertNerAdvanced_66159676227610
// MI455X (gfx1250) — compile-verified
//
#include <hip/hip_runtime.h>
#include <hip/hip_bf16.h>

// ---------------- problem constants ----------------
static constexpr int kB   = 64;
static constexpr int kS   = 512;
static constexpr int kH   = 768;
static constexpr int kW   = 256;
static constexpr int kLC  = 16;
static constexpr int kCE  = 30;
static constexpr int kNF  = 50;
static constexpr int kPE  = 25;
static constexpr int kHID = 256;
static constexpr int kNL  = 9;
static constexpr int kNLP = 16;                     // padded NL for WMMA tile
static constexpr int kCOMB  = kH + 3 * kNF + kPE;   // 943
static constexpr int kCOMBP = 944;                  // padded to multiple of 4 (WMMA K)
static constexpr int kROWS  = kB * kW;              // 16384 word rows
static constexpr int kLDA   = 964;                  // LDS stride for A panel (bank-friendly)

typedef __attribute__((ext_vector_type(2))) float v2f;
typedef __attribute__((ext_vector_type(8))) float v8f;
typedef __attribute__((ext_vector_type(4))) int   v4i;

#if defined(__gfx1250__) && __has_builtin(__builtin_amdgcn_global_load_async_to_lds_b128) && __has_builtin(__builtin_amdgcn_s_wait_asynccnt)
#define USE_ASYNC_LDS 1
typedef __attribute__((address_space(1))) v4i g_v4i;   // global <4 x i32>
typedef __attribute__((address_space(3))) v4i l_v4i;   // LDS <4 x i32>
#endif

// ---------------- 1) zero the combined buffer ----------------
__global__ void zero_combined(float4* __restrict__ p, int n4) {
    int i = blockIdx.x * blockDim.x + threadIdx.x;
    int stride = gridDim.x * blockDim.x;
    float4 z = make_float4(0.f, 0.f, 0.f, 0.f);
    for (; i < n4; i += stride) p[i] = z;
}

// ---------------- 2) scatter first-occurrence BERT rows ----------------
__global__ __launch_bounds__(192) void scatter_word_bert(
    const float* __restrict__ bert, const int* __restrict__ word_ids,
    float* __restrict__ combined) {
    int bs = blockIdx.x;
    int b = bs / kS, s = bs % kS;
    int wid  = word_ids[b * kS + s];
    int prev = (s == 0) ? -1 : word_ids[b * kS + s - 1];
    if (wid >= 0 && wid < kW && wid != prev) {
        const float4* src = (const float4*)(bert + ((size_t)b * kS + s) * kH);
        float4* dst = (float4*)(combined + ((size_t)b * kW + wid) * kCOMBP);
        dst[threadIdx.x] = src[threadIdx.x];   // 192 * 4 = 768 floats
    }
}

// ---------------- 3) char CNN (k = 2,3,4, 50 filters each, max-pool) --------
__global__ __launch_bounds__(160) void char_cnn(
    const int* __restrict__ char_ids, const float* __restrict__ char_emb,
    const float* __restrict__ w2, const float* __restrict__ b2,
    const float* __restrict__ w3, const float* __restrict__ b3,
    const float* __restrict__ w4, const float* __restrict__ b4,
    float* __restrict__ combined) {
    __shared__ float sE[kLC][kCE + 2];
    __shared__ float sW[3000 + 4500 + 6000];   // w2 | w3 | w4
    for (int i = threadIdx.x; i < 3000; i += blockDim.x) sW[i]        = w2[i];
    for (int i = threadIdx.x; i < 4500; i += blockDim.x) sW[3000 + i] = w3[i];
    for (int i = threadIdx.x; i < 6000; i += blockDim.x) sW[7500 + i] = w4[i];

    const int f = threadIdx.x;
    int k = 2; const float* wbase = nullptr; float bias = 0.f;
    if (f < 50)       { k = 2; wbase = &sW[f * 60];                 bias = b2[f]; }
    else if (f < 100) { k = 3; wbase = &sW[3000 + (f - 50) * 90];   bias = b3[f - 50]; }
    else if (f < 150) { k = 4; wbase = &sW[7500 + (f - 100) * 120]; bias = b4[f - 100]; }
    const int pad  = k / 2;
    const int lout = kLC + 2 * pad - k + 1;

    for (int wi = 0; wi < 8; ++wi) {
        int word = blockIdx.x * 8 + wi;
        __syncthreads();   // previous iteration done before overwriting sE
        for (int i = threadIdx.x; i < kLC * kCE; i += blockDim.x) {
            int t = i / kCE, c = i % kCE;
            int cid = char_ids[(size_t)word * kLC + t];
            sE[t][c] = char_emb[cid * kCE + c];
        }
        __syncthreads();
        if (f < 150) {
            float m = 0.0f;                       // relu output >= 0
            for (int t = 0; t < lout; ++t) {
                float s = bias;
                for (int dt = 0; dt < k; ++dt) {
                    int tp = t + dt - pad;
                    if (tp >= 0 && tp < kLC) {
                        #pragma unroll 6
                        for (int c = 0; c < kCE; ++c)
                            s += sE[tp][c] * wbase[c * k + dt];
                    }
                }
                s = s > 0.f ? s : 0.f;
                m = s > m ? s : m;
            }
            combined[(size_t)word * kCOMBP + kH + f] = m;
        }
    }
}

// ---------------- 4) position embedding fill ----------------
__global__ void pos_fill(const int* __restrict__ pos_ids,
                         const float* __restrict__ pos_emb,
                         float* __restrict__ combined) {
    int i = blockIdx.x * blockDim.x + threadIdx.x;
    if (i < kROWS * kPE) {
        int row = i / kPE, p = i % kPE;
        combined[(size_t)row * kCOMBP + kH + 3 * kNF + p] =
            pos_emb[pos_ids[row] * kPE + p];
    }
}

// ---------------- 5a) pad W_h -> W_hp[944][256] (row 943 zero) -------------
__global__ void pad_Wh(const float* __restrict__ W_h, float* __restrict__ W_hp) {
    int i = blockIdx.x * blockDim.x + threadIdx.x;
    if (i < kCOMBP * kHID) {
        int r = i / kHID, c = i % kHID;
        W_hp[i] = (r < kCOMB) ? W_h[r * kHID + c] : 0.f;
    }
}

// ---------------- 5b) pad W_c -> W_cp[256][16] (cols 9..15 zero) -----------
__global__ void pad_Wc(const float* __restrict__ W_c, float* __restrict__ W_cp) {
    int i = blockIdx.x * blockDim.x + threadIdx.x;
    if (i < kHID * kNLP) {
        int r = i / kNLP, c = i % kNLP;
        W_cp[i] = (c < kNL) ? W_c[r * kNL + c] : 0.f;
    }
}

// ---------------- 6) hidden = relu(combined @ W_hp + b_h) via f32 WMMA ------
// grid = 1024; block = 256 = 8 waves; wave w owns N-tile [w*32, w*32+32)
__global__ __launch_bounds__(256) void hidden_gemm_relu_wmma(
    const float* __restrict__ A,    // combined [16384][944]
    const float* __restrict__ Bm,   // W_hp [944][256] (pre-padded, no guards)
    const float* __restrict__ bias, // b_h [256]
    float* __restrict__ Hd) {       // hidden [16384][256]
    __shared__ float sA[16 * kLDA];

    const int mbase = blockIdx.x * 16;
    // stage 16 x 944 A-panel into LDS (async copy if toolchain exposes it)
    for (int i = threadIdx.x; i < 16 * (kCOMBP / 4); i += 256) {
        int r = i / (kCOMBP / 4), c4 = i % (kCOMBP / 4);
#if defined(USE_ASYNC_LDS)
        __builtin_amdgcn_global_load_async_to_lds_b128(
            (g_v4i*)(A + (size_t)(mbase + r) * kCOMBP + c4 * 4),
            (l_v4i*)&sA[r * kLDA + c4 * 4],
            0, 0);
#else
        ((float4*)&sA[r * kLDA])[c4] =
            ((const float4*)&A[(size_t)(mbase + r) * kCOMBP])[c4];
#endif
    }
#if defined(USE_ASYNC_LDS)
    __builtin_amdgcn_s_wait_asynccnt(0);
#endif
    __syncthreads();

    const int wave  = threadIdx.x >> 5;
    const int lane  = threadIdx.x & 31;
    const int nb0   = wave * 32;        // first 16-wide N tile, second at +16
    const int mrow  = lane & 15;        // A row for this lane
    const int khalf = lane >> 4;        // K-pair select (0 -> K {0,1}, 1 -> K {2,3})
    const int ncol  = lane & 15;        // B/C column

    const float* bp = Bm + nb0 + ncol;  // advances by kHID per K row
    __builtin_prefetch(bp, 0, 1);       // global_prefetch_b8 warm-up

    v8f acc0 = {}, acc1 = {};
    #pragma unroll 4
    for (int k0 = 0; k0 < kCOMBP; k0 += 4) {
        const int kr = k0 + 2 * khalf;
        v2f a, b0, b1;
        a.x  = sA[mrow * kLDA + kr + 0];
        a.y  = sA[mrow * kLDA + kr + 1];
        b0.x = bp[(size_t)(kr)     * kHID];
        b0.y = bp[(size_t)(kr + 1) * kHID];
        b1.x = bp[(size_t)(kr)     * kHID + 16];
        b1.y = bp[(size_t)(kr + 1) * kHID + 16];
        acc0 = __builtin_amdgcn_wmma_f32_16x16x4_f32(
            false, a, false, b0, (short)0, acc0, false, false);
        acc1 = __builtin_amdgcn_wmma_f32_16x16x4_f32(
            false, a, false, b1, (short)0, acc1, false, false);
    }

    const int c0 = nb0 + ncol, c1 = c0 + 16;
    const float bc0 = bias[c0], bc1 = bias[c1];
    #pragma unroll
    for (int i = 0; i < 8; ++i) {       // C layout: vgpr i -> row i + 8*khalf
        int m = mbase + i + 8 * khalf;
        float v0 = acc0[i] + bc0;
        float v1 = acc1[i] + bc1;
        Hd[(size_t)m * kHID + c0] = v0 > 0.f ? v0 : 0.f;
        Hd[(size_t)m * kHID + c1] = v1 > 0.f ? v1 : 0.f;
    }
}

// ---------------- 7) emissions = hidden @ W_cp + b_c via f32 WMMA ----------
// grid = 128; block = 256 = 8 waves; wave w owns M-tile blockIdx.x*128 + w*16
__global__ __launch_bounds__(256) void emissions_wmma(
    const float* __restrict__ Hd,   // hidden [16384][256]
    const float* __restrict__ Bm,   // W_cp [256][16] (padded)
    const float* __restrict__ bias, // b_c [9]
    float* __restrict__ out) {      // emissions [16384][9]
    __shared__ float sB[kHID * kNLP];   // 16 KB
    for (int i = threadIdx.x; i < kHID * kNLP; i += 256) sB[i] = Bm[i];
    __syncthreads();

    const int wave  = threadIdx.x >> 5;
    const int lane  = threadIdx.x & 31;
    const int mbase = blockIdx.x * 128 + wave * 16;
    const int mrow  = lane & 15;
    const int khalf = lane >> 4;
    const int ncol  = lane & 15;

    const float* ap = Hd + (size_t)(mbase + mrow) * kHID;

    v8f acc = {};
    #pragma unroll 8
    for (int k0 = 0; k0 < kHID; k0 += 4) {
        const int kr = k0 + 2 * khalf;
        v2f a, b;
        a.x = ap[kr + 0];
        a.y = ap[kr + 1];
        b.x = sB[(kr)     * kNLP + ncol];
        b.y = sB[(kr + 1) * kNLP + ncol];
        acc = __builtin_amdgcn_wmma_f32_16x16x4_f32(
            false, a, false, b, (short)0, acc, false, false);
    }

    if (ncol < kNL) {                   // drop padded columns
        const float bc = bias[ncol];
        #pragma unroll
        for (int i = 0; i < 8; ++i) {
            int m = mbase + i + 8 * khalf;
            out[(size_t)m * kNL + ncol] = acc[i] + bc;
        }
    }
}

// ---------------- launcher ----------------
extern "C" void kernel_launch(void* const* d_in, const int* in_sizes, int n_in,
                              void* d_out, int out_size, void* d_ws, size_t ws_size,
                              hipStream_t stream) {
    const float* bert     = (const float*)d_in[0];
    const int*   word_ids = (const int*)d_in[1];
    const int*   char_ids = (const int*)d_in[2];
    const int*   pos_ids  = (const int*)d_in[3];
    const float* char_emb = (const float*)d_in[4];
    const float* w2 = (const float*)d_in[5];  const float* b2 = (const float*)d_in[6];
    const float* w3 = (const float*)d_in[7];  const float* b3 = (const float*)d_in[8];
    const float* w4 = (const float*)d_in[9];  const float* b4 = (const float*)d_in[10];
    const float* pos_emb = (const float*)d_in[11];
    const float* W_h = (const float*)d_in[12]; const float* b_h = (const float*)d_in[13];
    const float* W_c = (const float*)d_in[14]; const float* b_c = (const float*)d_in[15];
    float* out = (float*)d_out;

    float* combined = (float*)d_ws;                         // 16384 x 944
    float* hidden   = combined + (size_t)kROWS * kCOMBP;    // 16384 x 256
    float* W_hp     = hidden   + (size_t)kROWS * kHID;      // 944 x 256
    float* W_cp     = W_hp     + (size_t)kCOMBP * kHID;     // 256 x 16

    int n4 = (kROWS * kCOMBP) / 4;
    zero_combined<<<2048, 256, 0, stream>>>((float4*)combined, n4);

    pad_Wh<<<(kCOMBP * kHID + 255) / 256, 256, 0, stream>>>(W_h, W_hp);
    pad_Wc<<<(kHID * kNLP + 255) / 256, 256, 0, stream>>>(W_c, W_cp);

    scatter_word_bert<<<kB * kS, 192, 0, stream>>>(bert, word_ids, combined);

    char_cnn<<<kROWS / 8, 160, 0, stream>>>(char_ids, char_emb,
                                            w2, b2, w3, b3, w4, b4, combined);

    pos_fill<<<(kROWS * kPE + 255) / 256, 256, 0, stream>>>(pos_ids, pos_emb, combined);

    hidden_gemm_relu_wmma<<<kROWS / 16, 256, 0, stream>>>(
        combined, W_hp, b_h, hidden);

    emissions_wmma<<<kROWS / 128, 256, 0, stream>>>(hidden, W_cp, b_c, out);
}